// GCN_27307402068411
// MI455X (gfx1250) — compile-verified
//
#include <hip/hip_runtime.h>

typedef __attribute__((ext_vector_type(2))) float v2f;
typedef __attribute__((ext_vector_type(8))) float v8f;

#define LN_EPS 1e-5f

// ---------------- utility kernels ----------------

__global__ void zero_kernel(float* __restrict__ p, long n) {
    long i = (long)blockIdx.x * blockDim.x + threadIdx.x;
    long stride = (long)gridDim.x * blockDim.x;
    for (; i < n; i += stride) p[i] = 0.0f;
}

__global__ void degree_kernel(const int* __restrict__ dst, float* __restrict__ deg, int E) {
    int e = blockIdx.x * blockDim.x + threadIdx.x;
    if (e < E) atomicAdd(&deg[dst[e]], 1.0f);
}

__global__ void invert_kernel(float* __restrict__ norm, int N) {
    int i = blockIdx.x * blockDim.x + threadIdx.x;
    if (i < N) {
        float d = norm[i];
        norm[i] = d > 0.0f ? 1.0f / d : 0.0f;
    }
}

// agg[dst] += x[src] * norm[dst]; one wave per edge, 4 features per lane (128 feats)
__global__ void scatter_kernel(const float* __restrict__ x, const int* __restrict__ src,
                               const int* __restrict__ dst, const float* __restrict__ norm,
                               float* __restrict__ agg, int E) {
    int t = blockIdx.x * blockDim.x + threadIdx.x;
    int e = t >> 5;
    int lane = t & 31;
    if (e >= E) return;
    int s = src[e];
    int d = dst[e];
    float nv = norm[d];
    const float4 xs = *(const float4*)(x + (long)s * 128 + lane * 4);
    float* ap = agg + (long)d * 128 + lane * 4;
    atomicAdd(ap + 0, xs.x * nv);
    atomicAdd(ap + 1, xs.y * nv);
    atomicAdd(ap + 2, xs.z * nv);
    atomicAdd(ap + 3, xs.w * nv);
}

// ---------------- fused SAGE layer GEMM ----------------
// y[i,:] = LN/ReLU( [x[i,:], agg[i,:]] @ W + b ),  K = 256 fixed.
// Block: COLT waves. Wave w owns output col-tile w (16 cols).
// Block covers G=8 row-groups of 16 rows; W staged ONCE per block into LDS
// (transposed, pad-68 stride -> conflict-free ds_load_b64 fragments),
// K processed in 4 chunks of 64 to bound LDS.
template <int COLT, bool LNRELU>
__global__ __launch_bounds__(COLT * 32) void sage_gemm_kernel(
    const float* __restrict__ x,    // [N,128]
    const float* __restrict__ agg,  // [N,128] (already normalized)
    const float* __restrict__ W,    // [256, COLT*16] row-major
    const float* __restrict__ bias, // [COLT*16]
    float* __restrict__ y,          // [N, COLT*16]
    int N) {
    constexpr int NDIM    = COLT * 16;
    constexpr int THREADS = COLT * 32;
    constexpr int G       = 8;   // 16-row groups per block
    constexpr int LDK     = 68;  // padded K stride (64 + 4): conflict-free b64

    __shared__ float Wt[NDIM * LDK];  // W chunk, transposed [n][kloc]
    __shared__ float As[16 * LDK];    // A tile [m][kloc]
    __shared__ float red_s[8 * 16];   // per-wave row sums
    __shared__ float red_q[8 * 16];   // per-wave row sums of squares

    const int tid  = threadIdx.x;
    const int wave = tid >> 5;  // owned col tile
    const int lane = tid & 31;
    const int ml   = lane & 15;
    const int half = lane >> 4;
    const long base = (long)blockIdx.x * (G * 16);

    v8f acc[G];
#pragma unroll
    for (int g = 0; g < G; ++g) acc[g] = v8f{};

    for (int kc = 0; kc < 4; ++kc) {
        const int k0 = kc * 64;
        __syncthreads();  // previous chunk fully consumed
        // ---- stage W chunk (transposed) : global coalesced -> LDS ----
        const float* wsrc = W + (long)k0 * NDIM;
        for (int idx = tid; idx < 64 * NDIM; idx += THREADS) {
            const int kloc = idx / NDIM;
            const int n    = idx - kloc * NDIM;
            Wt[n * LDK + kloc] = wsrc[idx];
        }
        const float* asrc = (k0 < 128) ? x : agg;
        const int koff = k0 & 127;
#pragma unroll
        for (int g = 0; g < G; ++g) {
            __syncthreads();  // Wt ready / previous As consumed
            // ---- stage 16x64 A tile (float4) ----
            for (int q = tid; q < 256; q += THREADS) {
                const int r  = q >> 4;
                const int kq = (q & 15) * 4;
                long rr = base + g * 16 + r;
                if (rr >= N) rr = N - 1;  // clamp: keep EXEC all-1s
                const float4 v = *(const float4*)(asrc + rr * 128 + koff + kq);
                *(float4*)&As[r * LDK + kq] = v;
            }
            __syncthreads();  // As ready
            v8f c = acc[g];
#pragma unroll
            for (int k = 0; k < 64; k += 4) {
                const v2f a = *(const v2f*)&As[ml * LDK + half * 2 + k];
                const v2f b = *(const v2f*)&Wt[(wave * 16 + ml) * LDK + half * 2 + k];
                c = __builtin_amdgcn_wmma_f32_16x16x4_f32(
                    false, a, false, b, (short)0, c, false, false);
            }
            acc[g] = c;
        }
    }

    // ---- epilogue: bias (+ LayerNorm + ReLU) + store ----
    const float bv = bias[wave * 16 + ml];
#pragma unroll
    for (int g = 0; g < G; ++g) {
        v8f c = acc[g];
#pragma unroll
        for (int r = 0; r < 8; ++r) c[r] += bv;

        if (LNRELU) {
            // per-wave partial row sums over this wave's 16 columns
#pragma unroll
            for (int r = 0; r < 8; ++r) {
                float sv = c[r];
                float qv = c[r] * c[r];
#pragma unroll
                for (int off = 1; off < 16; off <<= 1) {
                    sv += __shfl_xor(sv, off, 32);
                    qv += __shfl_xor(qv, off, 32);
                }
                if (ml == 0) {
                    red_s[wave * 16 + half * 8 + r] = sv;
                    red_q[wave * 16 + half * 8 + r] = qv;
                }
            }
            __syncthreads();
            if (tid < 16) {  // combine across the COLT waves
                float ts = 0.0f, tq = 0.0f;
#pragma unroll
                for (int w = 0; w < COLT; ++w) {
                    ts += red_s[w * 16 + tid];
                    tq += red_q[w * 16 + tid];
                }
                red_s[tid] = ts;
                red_q[tid] = tq;
            }
            __syncthreads();
#pragma unroll
            for (int r = 0; r < 8; ++r) {
                const int rowid = half * 8 + r;
                const float mu   = red_s[rowid] * (1.0f / NDIM);
                const float var  = red_q[rowid] * (1.0f / NDIM) - mu * mu;
                const float rstd = rsqrtf(var + LN_EPS);
                float v = (c[r] - mu) * rstd;
                c[r] = v > 0.0f ? v : 0.0f;
            }
            __syncthreads();  // protect red arrays before next group writes
        }

#pragma unroll
        for (int r = 0; r < 8; ++r) {
            const long ro = base + g * 16 + half * 8 + r;
            if (ro < N) y[ro * NDIM + wave * 16 + ml] = c[r];
        }
    }
}

// ---------------- launcher ----------------

extern "C" void kernel_launch(void* const* d_in, const int* in_sizes, int n_in,
                              void* d_out, int out_size, void* d_ws, size_t ws_size,
                              hipStream_t stream) {
    const float* h   = (const float*)d_in[0];
    const int*   src = (const int*)d_in[1];
    const int*   dst = (const int*)d_in[2];
    const float* W1  = (const float*)d_in[3];
    const float* b1  = (const float*)d_in[4];
    const float* W2  = (const float*)d_in[5];
    const float* b2  = (const float*)d_in[6];
    const float* W3  = (const float*)d_in[7];
    const float* b3  = (const float*)d_in[8];
    float* out = (float*)d_out;

    const int N = in_sizes[0] / 128;
    const int E = in_sizes[1];

    float* ws   = (float*)d_ws;
    float* norm = ws;                     // [N]
    float* agg  = norm + N;               // [N*128]
    float* h1   = agg + (size_t)N * 128;  // [N*128]
    float* h2   = h1  + (size_t)N * 128;  // [N*128]

    const long feat_n      = (long)N * 128;
    const int  gemm_blocks = (N + 127) / 128;  // 8 row-groups of 16 per block
    const int  scat_blocks = (int)(((long)E * 32 + 255) / 256);

    // normalization coefficients (1/in-degree)
    zero_kernel<<<1024, 256, 0, stream>>>(norm, N);
    degree_kernel<<<(E + 255) / 256, 256, 0, stream>>>(dst, norm, E);
    invert_kernel<<<(N + 255) / 256, 256, 0, stream>>>(norm, N);

    // layer 1: h1 = ReLU(LN( [h, agg(h)] @ W1 + b1 ))
    zero_kernel<<<2048, 256, 0, stream>>>(agg, feat_n);
    scatter_kernel<<<scat_blocks, 256, 0, stream>>>(h, src, dst, norm, agg, E);
    sage_gemm_kernel<8, true><<<gemm_blocks, 256, 0, stream>>>(h, agg, W1, b1, h1, N);

    // layer 2
    zero_kernel<<<2048, 256, 0, stream>>>(agg, feat_n);
    scatter_kernel<<<scat_blocks, 256, 0, stream>>>(h1, src, dst, norm, agg, E);
    sage_gemm_kernel<8, true><<<gemm_blocks, 256, 0, stream>>>(h1, agg, W2, b2, h2, N);

    // layer 3 (no LN / ReLU), 64 output features -> d_out
    zero_kernel<<<2048, 256, 0, stream>>>(agg, feat_n);
    scatter_kernel<<<scat_blocks, 256, 0, stream>>>(h2, src, dst, norm, agg, E);
    sage_gemm_kernel<4, false><<<gemm_blocks, 128, 0, stream>>>(h2, agg, W3, b3, out, N);
}